// TransportOperatorHeader_88991722373467
// MI455X (gfx1250) — compile-verified
//
#include <hip/hip_runtime.h>
#include <math.h>

// Problem dims (fixed by reference)
#define B_ 256
#define D_ 512
#define K_ 8
#define M_ 64
#define N_ 64

#define NT_TAYLOR 12       // Taylor terms for expm
#define S_SQUARE  3        // scaling-and-squaring steps (scale = 1/8)
#define EXP_SCALE 0.125f
#define LSP 66             // padded LDS row stride (64 + 2) to dodge bank conflicts

typedef float v2f __attribute__((ext_vector_type(2)));
typedef float v8f __attribute__((ext_vector_type(8)));

// ---------------------------------------------------------------------------
// Wave-level 64x64 (f32) matmul tile: each of 16 waves owns one 16x16 output
// tile, built from 16 V_WMMA_F32_16X16X4_F32 ops (K=64 in steps of 4), split
// into TWO independent accumulator chains (k<32, k>=32) to halve the WMMA
// RAW-dependence depth, then summed with 8 VALU adds.
// A/B/C live in LDS with row stride LSP. Caller handles barriers.
// VGPR layouts per CDNA5 ISA 7.12.2:
//   A 16x4 : lanes 0-15 rows, VGPR0 K={0|2}, VGPR1 K={1|3} (hi-half lanes +2)
//   B 4x16 : VGPR r holds rows r and r+2 (hi-half lanes), cols = lane%16
//   C 16x16: VGPR v holds rows v and v+8 (hi-half lanes), cols = lane%16
// ---------------------------------------------------------------------------
__device__ __forceinline__ void mm64(const float* __restrict__ A,
                                     const float* __restrict__ Bm,
                                     float* __restrict__ C,
                                     int lane, int wave) {
  const int ti = (wave >> 2) << 4;
  const int tj = (wave & 3) << 4;
  const int r  = lane & 15;
  const int hi = lane >> 4;          // 0 or 1
  v8f acc0 = {}, acc1 = {};
#pragma unroll
  for (int k0 = 0; k0 < 32; k0 += 4) {
    const int ka = k0 + (hi << 1);
    v2f a, b;
    a.x = A[(ti + r) * LSP + ka];
    a.y = A[(ti + r) * LSP + ka + 1];
    b.x = Bm[ka * LSP + tj + r];
    b.y = Bm[(ka + 1) * LSP + tj + r];
    acc0 = __builtin_amdgcn_wmma_f32_16x16x4_f32(false, a, false, b,
                                                 (short)0, acc0, false, false);
  }
#pragma unroll
  for (int k0 = 32; k0 < 64; k0 += 4) {
    const int ka = k0 + (hi << 1);
    v2f a, b;
    a.x = A[(ti + r) * LSP + ka];
    a.y = A[(ti + r) * LSP + ka + 1];
    b.x = Bm[ka * LSP + tj + r];
    b.y = Bm[(ka + 1) * LSP + tj + r];
    acc1 = __builtin_amdgcn_wmma_f32_16x16x4_f32(false, a, false, b,
                                                 (short)0, acc1, false, false);
  }
#pragma unroll
  for (int v = 0; v < 8; ++v)
    C[(ti + (hi << 3) + v) * LSP + tj + r] = acc0[v] + acc1[v];
}

// ---------------------------------------------------------------------------
// Kernel 1: T[b,k] = sum_m coef[b,m] * psi[k,m]  (written to Tg, unscaled)
//           E[b,k] = expm(T[b,k]) via Taylor(NT) + squaring(S) using WMMA.
// One workgroup (512 thr / 16 waves) per (b,k). 3 LDS tiles (~51 KB).
// ---------------------------------------------------------------------------
__global__ __launch_bounds__(512)
void k_buildT_expm(const float* __restrict__ coef,   // [B, M]
                   const float* __restrict__ psi,    // [K, M, 4096]
                   float* __restrict__ Tg,           // [B*K, 4096]
                   float* __restrict__ Eg)           // [B*K, 4096]
{
  __shared__ float yc[M_];
  __shared__ float A [64 * LSP];
  __shared__ float S [64 * LSP];
  __shared__ float Tm[64 * LSP];
  const int bk = blockIdx.x;
  const int b = bk >> 3, k = bk & 7;
  const int tid = threadIdx.x;
  const int lane = tid & 31, wave = tid >> 5;

  if (tid < M_) yc[tid] = coef[(size_t)b * M_ + tid];
  __syncthreads();

  // Accumulate T: each thread owns 8 flat elements e = i*512 + tid (coalesced)
  float acc[8];
#pragma unroll
  for (int i = 0; i < 8; ++i) acc[i] = 0.f;
  const float* pk = psi + (size_t)k * M_ * 4096;
  for (int m = 0; m < M_; ++m) {
    const float cm = yc[m];
    const float* pm = pk + (size_t)m * 4096;
    // hide L2 latency of the next psi matrix (emits global_prefetch_b8)
    __builtin_prefetch((const void*)(pm + 4096 + tid), 0, 3);
#pragma unroll
    for (int i = 0; i < 8; ++i)
      acc[i] = fmaf(cm, pm[i * 512 + tid], acc[i]);
  }

  float eReg[8];   // running exp accumulator (own elements) kept in registers
#pragma unroll
  for (int i = 0; i < 8; ++i) {
    const int e = i * 512 + tid;
    const int u = e >> 6, v = e & 63;
    Tg[(size_t)bk * 4096 + e] = acc[i];
    const float a = acc[i] * EXP_SCALE;
    A[u * LSP + v] = a;
    S[u * LSP + v] = a;
    eReg[i] = a + (u == v ? 1.f : 0.f);
  }
  __syncthreads();

  // Taylor: S_n = A*S_{n-1}/n ; E += S_n   (fully unrolled -> 1/n is literal)
#pragma unroll
  for (int n = 2; n <= NT_TAYLOR; ++n) {
    mm64(A, S, Tm, lane, wave);
    __syncthreads();
    const float invn = 1.f / (float)n;
#pragma unroll
    for (int i = 0; i < 8; ++i) {
      const int e = i * 512 + tid;
      const int u = e >> 6, v = e & 63;
      const float sn = Tm[u * LSP + v] * invn;
      S[u * LSP + v] = sn;
      eReg[i] += sn;
    }
    __syncthreads();
  }

  // Dump E into S (S_NT is dead), then square s times: E <- E*E
#pragma unroll
  for (int i = 0; i < 8; ++i) {
    const int e = i * 512 + tid;
    S[(e >> 6) * LSP + (e & 63)] = eReg[i];
  }
  __syncthreads();
#pragma unroll
  for (int j = 0; j < S_SQUARE; ++j) {
    mm64(S, S, Tm, lane, wave);
    __syncthreads();
#pragma unroll
    for (int i = 0; i < 8; ++i) {
      const int e = i * 512 + tid;
      S[(e >> 6) * LSP + (e & 63)] = Tm[(e >> 6) * LSP + (e & 63)];
    }
    __syncthreads();
  }
#pragma unroll
  for (int i = 0; i < 8; ++i) {
    const int e = i * 512 + tid;
    Eg[(size_t)bk * 4096 + e] = S[(e >> 6) * LSP + (e & 63)];
  }
}

// ---------------------------------------------------------------------------
// Kernel 2: r[b,k,u] = (E[b,k] @ z0_block)[u] - z1[b, k*64+u]
// ---------------------------------------------------------------------------
__global__ __launch_bounds__(64)
void k_residual(const float* __restrict__ Eg, const float* __restrict__ z0,
                const float* __restrict__ z1, float* __restrict__ rg) {
  const int bk = blockIdx.x;
  const int b = bk >> 3, k = bk & 7;
  const int u = threadIdx.x;
  const float* Er = Eg + (size_t)bk * 4096 + u * 64;
  const float* zb = z0 + (size_t)b * D_ + k * 64;
  float s = 0.f;
#pragma unroll 8
  for (int v = 0; v < 64; ++v) s = fmaf(Er[v], zb[v], s);
  rg[(size_t)bk * 64 + u] = s - z1[(size_t)b * D_ + k * 64 + u];
}

// ---------------------------------------------------------------------------
// Kernel 3: adjoint Fréchet derivative  Tbar = L_exp(T^T)[ r z0^T ]
// Coupled Taylor recurrence with rank-1 H (H*S = r (z0^T S), a matvec),
// then squaring chain:  L <- E*L + L*E ; E <- E*E.
// 5 LDS tiles (~85 KB of the 320 KB CDNA5 LDS) per workgroup.
// ---------------------------------------------------------------------------
__global__ __launch_bounds__(512)
void k_frechet(const float* __restrict__ Tg, const float* __restrict__ rg,
               const float* __restrict__ z0, float* __restrict__ Tbar) {
  __shared__ float A [64 * LSP];
  __shared__ float S [64 * LSP];
  __shared__ float F [64 * LSP];
  __shared__ float t1[64 * LSP];
  __shared__ float t2[64 * LSP];
  __shared__ float rv[64], zv[64], wv[64];
  const int bk = blockIdx.x;
  const int b = bk >> 3, k = bk & 7;
  const int tid = threadIdx.x;
  const int lane = tid & 31, wave = tid >> 5;

  if (tid < 64) {
    rv[tid] = rg[(size_t)bk * 64 + tid];
    zv[tid] = z0[(size_t)b * D_ + k * 64 + tid];
  }
  __syncthreads();

  float es[8], ls[8];
  // A = T^T / 2^s (transpose on LDS store); F = H = r z0^T
#pragma unroll
  for (int i = 0; i < 8; ++i) {
    const int e = i * 512 + tid;
    const int u = e >> 6, v = e & 63;
    const float a = Tg[(size_t)bk * 4096 + e] * EXP_SCALE;  // T[u][v]
    A[v * LSP + u] = a;                                     // A = scaled T^T
    const float h = rv[u] * zv[v];
    F[u * LSP + v] = h;
    ls[i] = h;                                              // L_1 = H
  }
  __syncthreads();
#pragma unroll
  for (int i = 0; i < 8; ++i) {
    const int e = i * 512 + tid;
    const int u = e >> 6, v = e & 63;
    const float a = A[u * LSP + v];
    S[u * LSP + v] = a;                                     // S_1 = A
    es[i] = a + (u == v ? 1.f : 0.f);                       // E = I + A
  }
  __syncthreads();

#pragma unroll
  for (int n = 2; n <= NT_TAYLOR; ++n) {
    mm64(A, S, t1, lane, wave);          // A * S_{n-1}
    mm64(A, F, t2, lane, wave);          // A * F_{n-1}
    if (tid < 64) {                      // w = z0^T S_{n-1}   (H*S is rank-1)
      float acc = 0.f;
#pragma unroll 8
      for (int u = 0; u < 64; ++u) acc = fmaf(zv[u], S[u * LSP + tid], acc);
      wv[tid] = acc;
    }
    __syncthreads();
    const float invn = 1.f / (float)n;
#pragma unroll
    for (int i = 0; i < 8; ++i) {
      const int e = i * 512 + tid;
      const int u = e >> 6, v = e & 63;
      const float sn = t1[u * LSP + v] * invn;
      S[u * LSP + v] = sn;  es[i] += sn;
      const float fn = (t2[u * LSP + v] + rv[u] * wv[v]) * invn;
      F[u * LSP + v] = fn;  ls[i] += fn;
    }
    __syncthreads();
  }

  // Materialize E -> S, L -> F for the squaring chain
#pragma unroll
  for (int i = 0; i < 8; ++i) {
    const int e = i * 512 + tid;
    const int u = e >> 6, v = e & 63;
    S[u * LSP + v] = es[i];
    F[u * LSP + v] = ls[i];
  }
  __syncthreads();
#pragma unroll
  for (int j = 0; j < S_SQUARE; ++j) {
    mm64(S, F, t1, lane, wave);          // E*L
    mm64(F, S, t2, lane, wave);          // L*E
    mm64(S, S, A,  lane, wave);          // E*E  (A is dead -> reuse)
    __syncthreads();
#pragma unroll
    for (int i = 0; i < 8; ++i) {
      const int e = i * 512 + tid;
      const int u = e >> 6, v = e & 63;
      F[u * LSP + v] = t1[u * LSP + v] + t2[u * LSP + v];
      S[u * LSP + v] = A[u * LSP + v];
    }
    __syncthreads();
  }
#pragma unroll
  for (int i = 0; i < 8; ++i) {
    const int e = i * 512 + tid;
    Tbar[(size_t)bk * 4096 + e] = F[(e >> 6) * LSP + (e & 63)];
  }
}

// ---------------------------------------------------------------------------
// Kernel 4: g[b,m] = sum_{k,u,v} Tbar[b,k,u,v] * psi[k,m,u,v]
// One workgroup per (b,m); coalesced reads, LDS tree reduction.
// ---------------------------------------------------------------------------
__global__ __launch_bounds__(256)
void k_grad(const float* __restrict__ Tbar, const float* __restrict__ psi,
            float* __restrict__ g) {
  __shared__ float red[256];
  const int bm = blockIdx.x;
  const int b = bm >> 6, m = bm & 63;
  const int tid = threadIdx.x;
  float acc = 0.f;
  const float* tb = Tbar + (size_t)b * (K_ * 4096);
  for (int idx = tid; idx < K_ * 4096; idx += 256) {
    const int k = idx >> 12, uv = idx & 4095;
    acc = fmaf(tb[idx],
               psi[(size_t)k * (M_ * 4096) + (size_t)m * 4096 + uv], acc);
  }
  red[tid] = acc;
  __syncthreads();
  for (int off = 128; off > 0; off >>= 1) {
    if (tid < off) red[tid] += red[tid + off];
    __syncthreads();
  }
  if (tid == 0) g[(size_t)b * M_ + m] = red[0];
}

// ---------------------------------------------------------------------------
// Kernel 5: FISTA step. c_new = shrink(y - lr*g, thr);
//           y = c_new + beta*(c_new - c_old); c = c_new.
// ---------------------------------------------------------------------------
__global__ __launch_bounds__(256)
void k_update(const float* __restrict__ g, float* __restrict__ c,
              float* __restrict__ y, float lr, float thr, float beta) {
  const int i = blockIdx.x * 256 + threadIdx.x;
  if (i >= B_ * M_) return;
  const float cp = c[i];
  const float x  = y[i] - lr * g[i];
  const float mag = fabsf(x) - thr;
  const float cn = (mag > 0.f ? mag : 0.f) * (x > 0.f ? 1.f : (x < 0.f ? -1.f : 0.f));
  c[i] = cn;
  y[i] = cn + beta * (cn - cp);
}

__global__ __launch_bounds__(256)
void k_zero(float* __restrict__ p, int n) {
  const int i = blockIdx.x * 256 + threadIdx.x;
  if (i < n) p[i] = 0.f;
}

// Kernel 6: final z1_hat = E @ z0 -> d_out
__global__ __launch_bounds__(64)
void k_final(const float* __restrict__ Eg, const float* __restrict__ z0,
             float* __restrict__ out) {
  const int bk = blockIdx.x;
  const int b = bk >> 3, k = bk & 7;
  const int u = threadIdx.x;
  const float* Er = Eg + (size_t)bk * 4096 + u * 64;
  const float* zb = z0 + (size_t)b * D_ + k * 64;
  float s = 0.f;
#pragma unroll 8
  for (int v = 0; v < 64; ++v) s = fmaf(Er[v], zb[v], s);
  out[(size_t)b * D_ + k * 64 + u] = s;
}

extern "C" void kernel_launch(void* const* d_in, const int* in_sizes, int n_in,
                              void* d_out, int out_size, void* d_ws, size_t ws_size,
                              hipStream_t stream) {
  (void)in_sizes; (void)n_in; (void)out_size; (void)ws_size;
  const float* z0  = (const float*)d_in[0];   // feature_0 [B, D]
  const float* z1  = (const float*)d_in[1];   // feature_1 [B, D]
  const float* psi = (const float*)d_in[2];   // psi [K, M, N, N]

  char* ws = (char*)d_ws;
  const size_t mat_bytes = (size_t)B_ * K_ * 4096 * sizeof(float);   // 33.5 MB
  float* Tg = (float*)ws;  ws += mat_bytes;
  float* Eg = (float*)ws;  ws += mat_bytes;                          // E, later Tbar
  float* rg = (float*)ws;  ws += (size_t)B_ * K_ * 64 * sizeof(float);
  float* c  = (float*)ws;  ws += (size_t)B_ * M_ * sizeof(float);
  float* y  = (float*)ws;  ws += (size_t)B_ * M_ * sizeof(float);
  float* g  = (float*)ws;  ws += (size_t)B_ * M_ * sizeof(float);

  const int BM = B_ * M_;
  k_zero<<<(2 * BM + 255) / 256, 256, 0, stream>>>(c, 2 * BM);  // c,y contiguous

  const float lr  = 0.01f;
  const float thr = 0.01f * 0.05f;   // FISTA_LR * LAMBDA_PRIOR
  double t = 1.0;                     // momentum sequence is input-independent
  for (int it = 0; it < 20; ++it) {
    k_buildT_expm<<<B_ * K_, 512, 0, stream>>>(y, psi, Tg, Eg);
    k_residual  <<<B_ * K_,  64, 0, stream>>>(Eg, z0, z1, rg);
    k_frechet   <<<B_ * K_, 512, 0, stream>>>(Tg, rg, z0, Eg);   // Tbar -> Eg
    k_grad      <<<B_ * M_, 256, 0, stream>>>(Eg, psi, g);
    const double tn = 0.5 * (1.0 + sqrt(1.0 + 4.0 * t * t));
    const float beta = (float)((t - 1.0) / tn);
    t = tn;
    k_update<<<(BM + 255) / 256, 256, 0, stream>>>(g, c, y, lr, thr, beta);
  }
  // Final transport with inferred c
  k_buildT_expm<<<B_ * K_, 512, 0, stream>>>(c, psi, Tg, Eg);
  k_final      <<<B_ * K_,  64, 0, stream>>>(Eg, z0, (float*)d_out);
}